// DKNN_19756849562155
// MI455X (gfx1250) — compile-verified
//
#include <hip/hip_runtime.h>
#include <hip/hip_bf16.h>

typedef __attribute__((ext_vector_type(2))) float v2f;
typedef __attribute__((ext_vector_type(8))) float v8f;

#define MDIM 2048
#define NDIM 4096
#define KDIM 256
#define ITERS 200
#define COLS 16
#define TPB  256

#define WMMA_F32(A, B, C) \
  __builtin_amdgcn_wmma_f32_16x16x4_f32(false, (A), false, (B), (short)0, (C), false, false)

// ---------------------------------------------------------------------------
// 0. Workspace init: ws[0] holds global max(|cos|, |1-cos|) as uint-ordered f32.
// ---------------------------------------------------------------------------
__global__ void dknn_init_ws(unsigned* ws) {
  if (threadIdx.x == 0) ws[0] = 0u;
}

// ---------------------------------------------------------------------------
// 1. cos = Q @ N^T via V_WMMA_F32_16X16X4_F32.
//    Each wave computes a 32(M) x 64(N) strip: 2 A-frags x 4 B-frags -> 8 WMMAs
//    per 6 b64 loads. Register double-buffering: K-step k0+4 fragments are
//    loaded before the k0 WMMAs issue, so s_wait_loadcnt overlaps with matrix
//    work instead of stalling before every WMMA. Uniform base + 32-bit VGPR
//    offsets -> saddr global loads (no per-lane 64-bit pointer chains).
//    32-bit A 16x4 layout: lane=(half<<4)|r ; M=r ; K=vgpr+2*half
//    B 4x16 mirrored:      N=r ; K=vgpr+2*half ; B^T[k][n]=Nb[n][k]
//    Fused epilogue: distances(=cos), scores(=1-cos), global max(|cos|,|s|).
// ---------------------------------------------------------------------------
__global__ __launch_bounds__(128) void dknn_gemm_wmma(
    const float* __restrict__ Q, const float* __restrict__ Nb,
    float* __restrict__ dist, float* __restrict__ scor,
    unsigned* __restrict__ wsmax)
{
  const int lane = threadIdx.x & 31;
  const int wave = threadIdx.x >> 5;
  const int half = lane >> 4;
  const int r    = lane & 15;
  const int tileM = blockIdx.y * 32;
  const int tileN = blockIdx.x * 256 + wave * 64;

  // 32-bit element offsets (uniform base pointer + divergent offset -> saddr)
  const int aOff0 = (tileM      + r) * KDIM + 2 * half;
  const int aOff1 = aOff0 + 16 * KDIM;
  const int bOff0 = (tileN      + r) * KDIM + 2 * half;
  const int bOff1 = bOff0 + 16 * KDIM;
  const int bOff2 = bOff0 + 32 * KDIM;
  const int bOff3 = bOff0 + 48 * KDIM;

  v8f acc[8] = {};

  // prologue: K-step 0 fragments
  v2f a0 = *(const v2f*)(Q  + aOff0);
  v2f a1 = *(const v2f*)(Q  + aOff1);
  v2f b0 = *(const v2f*)(Nb + bOff0);
  v2f b1 = *(const v2f*)(Nb + bOff1);
  v2f b2 = *(const v2f*)(Nb + bOff2);
  v2f b3 = *(const v2f*)(Nb + bOff3);

  #pragma unroll 2
  for (int k0 = 4; k0 < KDIM; k0 += 4) {
    // issue next K-step loads before consuming current fragments
    v2f a0n = *(const v2f*)(Q  + aOff0 + k0);
    v2f a1n = *(const v2f*)(Q  + aOff1 + k0);
    v2f b0n = *(const v2f*)(Nb + bOff0 + k0);
    v2f b1n = *(const v2f*)(Nb + bOff1 + k0);
    v2f b2n = *(const v2f*)(Nb + bOff2 + k0);
    v2f b3n = *(const v2f*)(Nb + bOff3 + k0);

    acc[0] = WMMA_F32(a0, b0, acc[0]);
    acc[1] = WMMA_F32(a0, b1, acc[1]);
    acc[2] = WMMA_F32(a0, b2, acc[2]);
    acc[3] = WMMA_F32(a0, b3, acc[3]);
    acc[4] = WMMA_F32(a1, b0, acc[4]);
    acc[5] = WMMA_F32(a1, b1, acc[5]);
    acc[6] = WMMA_F32(a1, b2, acc[6]);
    acc[7] = WMMA_F32(a1, b3, acc[7]);

    a0 = a0n; a1 = a1n; b0 = b0n; b1 = b1n; b2 = b2n; b3 = b3n;
  }
  // epilogue K-step
  acc[0] = WMMA_F32(a0, b0, acc[0]);
  acc[1] = WMMA_F32(a0, b1, acc[1]);
  acc[2] = WMMA_F32(a0, b2, acc[2]);
  acc[3] = WMMA_F32(a0, b3, acc[3]);
  acc[4] = WMMA_F32(a1, b0, acc[4]);
  acc[5] = WMMA_F32(a1, b1, acc[5]);
  acc[6] = WMMA_F32(a1, b2, acc[6]);
  acc[7] = WMMA_F32(a1, b3, acc[7]);

  // D layout (32-bit C/D 16x16): lane n = r, VGPR i -> row M = i + 8*half
  float lmax = 0.f;
  #pragma unroll
  for (int mt = 0; mt < 2; ++mt) {
    #pragma unroll
    for (int nt = 0; nt < 4; ++nt) {
      const v8f av = acc[mt * 4 + nt];
      const int n  = tileN + nt * 16 + r;
      #pragma unroll
      for (int i = 0; i < 8; ++i) {
        const int row = tileM + mt * 16 + i + 8 * half;
        const float c = av[i];
        const float s = 1.0f - c;
        const size_t idx = (size_t)row * NDIM + n;
        dist[idx] = c;
        scor[idx] = s;
        lmax = fmaxf(lmax, fmaxf(fabsf(c), fabsf(s)));
      }
    }
  }

  // block max -> global atomic max (nonneg floats: uint order == float order)
  #pragma unroll
  for (int o = 16; o > 0; o >>= 1)
    lmax = fmaxf(lmax, __shfl_xor(lmax, o, 32));
  __shared__ float smax[4];
  if (lane == 0) smax[wave] = lmax;
  __syncthreads();
  if (threadIdx.x == 0) {
    float m = fmaxf(fmaxf(smax[0], smax[1]), fmaxf(smax[2], smax[3]));
    atomicMax(wsmax, __float_as_uint(m));
  }
}

// ---------------------------------------------------------------------------
// 2. Per-row Sinkhorn, multiplicative form (mathematically identical to the
//    reference's stabilized log-domain updates; f32-safe for this data range):
//      K_k = exp(-C_k/eps)      (precomputed once per column, 2 exps total)
//      u   = mu  / (v0*K0 + v1*K1)        per column
//      v_k = nu_k / sum_j u_j*K_kj        block reduction
//    Gamma[:, :, 0] = u * K0 * v0 ; top = Gamma0 * n, thresholded at 0.3.
//    One block per query row; 16 columns/thread kept in registers.
// ---------------------------------------------------------------------------
__global__ __launch_bounds__(TPB) void dknn_sinkhorn(
    const float* __restrict__ scor, float* __restrict__ top,
    const unsigned* __restrict__ wsmax)
{
  const int row = blockIdx.x;
  const int t   = threadIdx.x;
  __shared__ float shx[8], shy[8];

  const float m    = __uint_as_float(wsmax[0]);
  const float invc = __builtin_amdgcn_rcpf(m * m);   // 1 / C.max()

  const float* srow = scor + (size_t)row * NDIM;
  float K0[COLS], K1[COLS];
  #pragma unroll
  for (int c = 0; c < COLS; ++c) {
    const float s  = srow[c * TPB + t];
    const float C0 = s * s * invc;                 // (s-0)^2 / Cmax
    const float sm1 = s - 1.0f;
    const float C1 = sm1 * sm1 * invc;             // (s-1)^2 / Cmax
    K0[c] = __expf(-10.0f * C0);                   // exp(-C/eps), eps=0.1
    K1[c] = __expf(-10.0f * C1);
  }

  const float MU  = 1.0f / 4096.0f;     // mu
  const float NU0 = 16.0f / 4096.0f;    // K/n
  const float NU1 = 4080.0f / 4096.0f;  // (n-K)/n
  float v0 = 1.0f, v1 = 1.0f;           // g = 0 initially

  for (int it = 0; it < ITERS; ++it) {
    float t0 = 0.f, t1 = 0.f;
    #pragma unroll
    for (int c = 0; c < COLS; ++c) {
      const float d = fmaf(v0, K0[c], v1 * K1[c]);
      const float u = MU * __builtin_amdgcn_rcpf(d);   // u_j = mu/(Kv)_j
      t0 = fmaf(u, K0[c], t0);
      t1 = fmaf(u, K1[c], t1);
    }
    // block sum of (t0,t1): wave32 shuffle + 8-leader LDS tree
    #pragma unroll
    for (int o = 16; o > 0; o >>= 1) {
      t0 += __shfl_xor(t0, o, 32);
      t1 += __shfl_xor(t1, o, 32);
    }
    const int wid = t >> 5;
    if ((t & 31) == 0) { shx[wid] = t0; shy[wid] = t1; }
    __syncthreads();
    float S0 = 0.f, S1 = 0.f;
    #pragma unroll
    for (int w = 0; w < 8; ++w) { S0 += shx[w]; S1 += shy[w]; }
    __syncthreads();                       // LDS reused next iteration
    v0 = NU0 * __builtin_amdgcn_rcpf(S0);  // v_k = nu_k / (K^T u)_k
    v1 = NU1 * __builtin_amdgcn_rcpf(S1);
  }

  // reference does one extra f-update (u-update) before Gamma = u K v
  float* orow = top + (size_t)row * NDIM;
  #pragma unroll
  for (int c = 0; c < COLS; ++c) {
    const float d  = fmaf(v0, K0[c], v1 * K1[c]);
    const float u  = MU * __builtin_amdgcn_rcpf(d);
    float tk = u * K0[c] * v0 * 4096.0f;   // Gamma[:, :, 0] * n
    if (tk < 0.3f) tk = 0.0f;
    orow[c * TPB + t] = tk;
  }
}

// ---------------------------------------------------------------------------
extern "C" void kernel_launch(void* const* d_in, const int* in_sizes, int n_in,
                              void* d_out, int out_size, void* d_ws, size_t ws_size,
                              hipStream_t stream) {
  const float* Q  = (const float*)d_in[0];   // (2048, 256)
  const float* Nb = (const float*)d_in[1];   // (4096, 256)
  float* out  = (float*)d_out;
  float* top  = out;                              // (2048, 4096)
  float* dist = out + (size_t)MDIM * NDIM;        // distances = cos
  float* scor = out + 2 * (size_t)MDIM * NDIM;    // scores    = 1 - cos
  unsigned* wsmax = (unsigned*)d_ws;

  hipLaunchKernelGGL(dknn_init_ws, dim3(1), dim3(32), 0, stream, wsmax);
  hipLaunchKernelGGL(dknn_gemm_wmma, dim3(NDIM / 256, MDIM / 32), dim3(128), 0, stream,
                     Q, Nb, dist, scor, wsmax);
  hipLaunchKernelGGL(dknn_sinkhorn, dim3(MDIM), dim3(TPB), 0, stream,
                     scor, top, wsmax);
}